// EquivariantLayerNorm_25108378812473
// MI455X (gfx1250) — compile-verified
//
#include <hip/hip_runtime.h>
#include <hip/hip_bf16.h>

// EquivariantLayerNorm for irreps = 128x0e + 64x1o + 32x2e (dim = 480).
// Memory-bound: ~384 MB traffic -> ~16.5 us roofline @ 23.3 TB/s.
// CDNA5 path: async global->LDS staging (ASYNCcnt) + wave32 shuffle reductions.
// Steady state per row: 4x global_load_async_to_lds_b128, ~12 ds loads,
// pure-FMA affine (expanded w_full built once per block), 4x b128 stores.

#define WAVES_PER_BLOCK 8
#define BLOCK_THREADS   256
#define ROW   480        // floats per row
#define ROWB  1920       // bytes per row
#define SM_WFULL 7680    // float index of w_full[480] in smem
#define SM_BIAS  8160    // float index of bias[128] in smem
#define SM_TOTAL 8288    // 8 waves * 2 buffers * 480 + 480 + 128

__device__ __forceinline__ float waveSum(float v) {
#pragma unroll
  for (int o = 16; o > 0; o >>= 1) v += __shfl_xor(v, o, 32);
  return v;
}

// Issue one row (1920 B) as 4 async b128 transfers: 3 full-wave + 1 with 24 lanes.
// INST_OFFSET is added to BOTH the global and the LDS address (ISA 08 S4.4),
// so a single base pair covers the whole row.
__device__ __forceinline__ void issueRow(const float* g, unsigned ldsByte, int lane) {
  const float* ga = g + lane * 4;               // 16 B per lane
  unsigned la = ldsByte + (unsigned)lane * 16u; // LDS byte address (array at LDS base 0)
  asm volatile("global_load_async_to_lds_b128 %0, %1, off"             :: "v"(la), "v"(ga) : "memory");
  asm volatile("global_load_async_to_lds_b128 %0, %1, off offset:512"  :: "v"(la), "v"(ga) : "memory");
  asm volatile("global_load_async_to_lds_b128 %0, %1, off offset:1024" :: "v"(la), "v"(ga) : "memory");
  if (lane < 24) // 480 floats = 3*128 + 96 -> last op covers 24 lanes * 4 floats
    asm volatile("global_load_async_to_lds_b128 %0, %1, off offset:1536" :: "v"(la), "v"(ga) : "memory");
}

__global__ __launch_bounds__(BLOCK_THREADS) void eln_kernel(const float* __restrict__ in,
                                                            const float* __restrict__ w,
                                                            const float* __restrict__ b,
                                                            float* __restrict__ out, int n) {
  __shared__ float smem[SM_TOTAL];

  // ---- one-time per block: expand per-instance weight to per-channel w_full[480] ----
  // (the /3 and /5 happen here once, never in the row loop)
#pragma unroll
  for (int t = 0; t < 2; ++t) {
    int i = threadIdx.x + t * BLOCK_THREADS;
    if (i < ROW) {
      float wv;
      if (i < 128)      wv = w[i];
      else if (i < 320) wv = w[128 + (i - 128) / 3];
      else              wv = w[192 + (i - 320) / 5];
      smem[SM_WFULL + i] = wv;
    }
  }
  if (threadIdx.x < 128) smem[SM_BIAS + threadIdx.x] = b[threadIdx.x];
  __syncthreads();

  const int lane   = threadIdx.x & 31;
  const int wv     = threadIdx.x >> 5;
  const int gw     = blockIdx.x * WAVES_PER_BLOCK + wv;
  const int stride = gridDim.x * WAVES_PER_BLOCK;

  // Row-invariant per-lane affine constants, hoisted to registers.
  const float4 wf0 = *(const float4*)(smem + SM_WFULL +   0 + lane * 4);
  const float4 wf1 = *(const float4*)(smem + SM_WFULL + 128 + lane * 4);
  const float4 wf2 = *(const float4*)(smem + SM_WFULL + 256 + lane * 4);
  const float4 wf3 = (lane < 24) ? *(const float4*)(smem + SM_WFULL + 384 + lane * 4)
                                 : make_float4(0.f, 0.f, 0.f, 0.f);
  const float4 bs  = *(const float4*)(smem + SM_BIAS + lane * 4);

  const float e  = 1e-6f;
  const float e2 = 1e-12f; // eps^2

  int cur = 0;
  if (gw < n) issueRow(in + (size_t)gw * ROW, (unsigned)(wv * 2 + 0) * ROWB, lane);

  for (int row = gw; row < n; row += stride) {
    const int nxt = row + stride;
    if (nxt < n) {
      // Prefetch next row into the other buffer, then drain only the 4 older ops.
      issueRow(in + (size_t)nxt * ROW, (unsigned)(wv * 2 + (cur ^ 1)) * ROWB, lane);
      asm volatile("s_wait_asynccnt 0x4" ::: "memory");
    } else {
      asm volatile("s_wait_asynccnt 0x0" ::: "memory");
    }

    const float* buf = smem + (wv * 2 + cur) * ROW;

    // ---- 1) mean of the 128 scalar (0e) channels (lane owns 4 contiguous) ----
    const float4 s4 = *(const float4*)(buf + lane * 4);
    const float mean = waveSum(s4.x + s4.y + s4.z + s4.w) * (1.0f / 128.0f);

    // ---- 2) per-instance norms, accumulate norm^2 (224 instances / 32 lanes) ----
    float acc = 0.0f;
    {                                           // 4 scalar instances per lane (d=1)
      float sc[4] = {s4.x, s4.y, s4.z, s4.w};
#pragma unroll
      for (int j = 0; j < 4; ++j) {
        float nn = fabsf(sc[j] - mean + e2) - e;
        acc += nn * nn;
      }
    }
#pragma unroll
    for (int t = 0; t < 2; ++t) {               // 2 l=1 instances per lane (d=3)
      const float* p = buf + 128 + (lane * 2 + t) * 3;
      float a0 = p[0] + e2, a1 = p[1] + e2, a2 = p[2] + e2;
      float nn = sqrtf(a0 * a0 + a1 * a1 + a2 * a2) - e;
      acc += nn * nn;
    }
    {                                           // 1 l=2 instance per lane (d=5)
      const float* p = buf + 320 + lane * 5;
      float a0 = p[0] + e2, a1 = p[1] + e2, a2 = p[2] + e2, a3 = p[3] + e2, a4 = p[4] + e2;
      float nn = sqrtf(a0 * a0 + a1 * a1 + a2 * a2 + a3 * a3 + a4 * a4) - e;
      acc += nn * nn;
    }

    // ---- 3) RMS over 224 instances ----
    const float inv_rms = rsqrtf(waveSum(acc) * (1.0f / 224.0f));

    // ---- 4/5) affine scale (+bias on scalars), coalesced b128 stores ----
    float* orow = out + (size_t)row * ROW;
    {
      float4 r;  // k = 0: scalar channels, reuse s4
      r.x = (s4.x - mean) * inv_rms * wf0.x + bs.x;
      r.y = (s4.y - mean) * inv_rms * wf0.y + bs.y;
      r.z = (s4.z - mean) * inv_rms * wf0.z + bs.z;
      r.w = (s4.w - mean) * inv_rms * wf0.w + bs.w;
      *(float4*)(orow + lane * 4) = r;
    }
    {
      const int c0 = 128 + lane * 4;
      const float4 v4 = *(const float4*)(buf + c0);
      float4 r;
      r.x = v4.x * inv_rms * wf1.x;
      r.y = v4.y * inv_rms * wf1.y;
      r.z = v4.z * inv_rms * wf1.z;
      r.w = v4.w * inv_rms * wf1.w;
      *(float4*)(orow + c0) = r;
    }
    {
      const int c0 = 256 + lane * 4;
      const float4 v4 = *(const float4*)(buf + c0);
      float4 r;
      r.x = v4.x * inv_rms * wf2.x;
      r.y = v4.y * inv_rms * wf2.y;
      r.z = v4.z * inv_rms * wf2.z;
      r.w = v4.w * inv_rms * wf2.w;
      *(float4*)(orow + c0) = r;
    }
    if (lane < 24) {
      const int c0 = 384 + lane * 4;
      const float4 v4 = *(const float4*)(buf + c0);
      float4 r;
      r.x = v4.x * inv_rms * wf3.x;
      r.y = v4.y * inv_rms * wf3.y;
      r.z = v4.z * inv_rms * wf3.z;
      r.w = v4.w * inv_rms * wf3.w;
      *(float4*)(orow + c0) = r;
    }
    cur ^= 1;
  }
}

extern "C" void kernel_launch(void* const* d_in, const int* in_sizes, int n_in,
                              void* d_out, int out_size, void* d_ws, size_t ws_size,
                              hipStream_t stream) {
  const float* in = (const float*)d_in[0];   // [N,480]
  const float* w  = (const float*)d_in[1];   // [224]
  const float* b  = (const float*)d_in[2];   // [128]
  float* out = (float*)d_out;                // [N,480]
  const int n = in_sizes[0] / ROW;

  int blocks = (n + WAVES_PER_BLOCK - 1) / WAVES_PER_BLOCK;
  if (blocks > 4096) blocks = 4096;
  if (blocks < 1) blocks = 1;
  eln_kernel<<<blocks, BLOCK_THREADS, 0, stream>>>(in, w, b, out, n);
}